// KLRS_87290915324268
// MI455X (gfx1250) — compile-verified
//
#include <hip/hip_runtime.h>

// CBCE loss for MI455X (gfx1250, wave32).
// Memory-bound: 524 MB single streaming read -> ~22.5us roofline @ 23.3 TB/s.
// - Tile staging uses CDNA5 async global->LDS DMA (GLOBAL_LOAD_ASYNC_TO_LDS_B128,
//   ASYNCcnt), bypassing VGPRs for deeper outstanding-load pipelining.
// - Row-sum of exp() runs on the matrix pipe via V_WMMA_F32_16X16X4_F32
//   (A = 16x4 exp-tile fragment, B = ones -> every D column = 16 row sums).

#define NUM_CLASSES 1000
#define ROW_STRIDE 1004               // LDS stride (floats); r*1004 mod 64 distinct mult-of-4 -> conflict-free b64 reads
#define ROWS_PER_WAVE 16
#define WAVES_PER_BLOCK 4
#define BLOCK_THREADS (WAVES_PER_BLOCK * 32)
#define ROWS_PER_BLOCK (WAVES_PER_BLOCK * ROWS_PER_WAVE)   // 64 rows, tile = 64*1004*4 = 257,024 B LDS (<320KB)

typedef float v2f __attribute__((ext_vector_type(2)));
typedef float v8f __attribute__((ext_vector_type(8)));

// Async DMA: global 16B -> LDS, no VGPR round-trip. Tracked by ASYNCcnt.
__device__ __forceinline__ void async_copy_b128(unsigned int lds_byte_off,
                                                unsigned long long gaddr)
{
    asm volatile("global_load_async_to_lds_b128 %0, %1, off"
                 :: "v"(lds_byte_off), "v"(gaddr)
                 : "memory");
}

__device__ __forceinline__ void wait_asynccnt0()
{
    asm volatile("s_wait_asynccnt 0x0" ::: "memory");
}

__global__ __launch_bounds__(BLOCK_THREADS)
void cbce_rows_kernel(const float* __restrict__ x,
                      const long long* __restrict__ target,
                      const float* __restrict__ cls_w,
                      float* __restrict__ sums,
                      float* __restrict__ counts,
                      int n_rows)
{
    __shared__ float tile[ROWS_PER_BLOCK * ROW_STRIDE];
    __shared__ float rowsum[WAVES_PER_BLOCK][ROWS_PER_WAVE];

    const int tid  = threadIdx.x;
    const int lane = tid & 31;
    const int wave = tid >> 5;
    const int lo   = lane & 15;        // row within wave tile (A-fragment M)
    const int hi   = lane >> 4;        // selects K pair {0,1} vs {2,3}

    const long long rowBase = (long long)blockIdx.x * ROWS_PER_BLOCK;
    const float* gsrc = x + rowBase * NUM_CLASSES;
    const long long totalElems = (long long)n_rows * NUM_CLASSES;

    // ---- Stage tile into LDS: contiguous 16B async DMA (perfectly coalesced) ----
    const int f4PerRow = NUM_CLASSES / 4;             // 250
    const int totF4    = ROWS_PER_BLOCK * f4PerRow;   // 16000
    for (int i = tid; i < totF4; i += BLOCK_THREADS) {
        const int r  = i / f4PerRow;
        const int c4 = i - r * f4PerRow;
        const long long g = rowBase * NUM_CLASSES + (long long)i * 4;
        if (g + 3 < totalElems) {
            const unsigned int lds_off =
                (unsigned int)(unsigned long long)(uintptr_t)&tile[r * ROW_STRIDE + (c4 << 2)];
            async_copy_b128(lds_off, (unsigned long long)(uintptr_t)(((const float4*)gsrc) + i));
        }
    }
    wait_asynccnt0();        // this wave's DMAs landed in LDS
    __syncthreads();         // all waves' tiles visible

    // ---- Pass 1: per-row max (A-fragment access pattern, bank-conflict-free b64) ----
    const float* myrow = &tile[(wave * ROWS_PER_WAVE + lo) * ROW_STRIDE];
    float mx = -3.402823e38f;
    for (int k0 = 0; k0 < NUM_CLASSES; k0 += 4) {
        const float2 v = *(const float2*)&myrow[k0 + (hi << 1)];
        mx = fmaxf(mx, fmaxf(v.x, v.y));
    }
    mx = fmaxf(mx, __shfl_xor(mx, 16, 32));           // combine lane lo with lane lo+16

    // ---- Pass 2: sum_k exp(x - m) via WMMA f32 16x16x4, B = ones ----
    v8f acc = (v8f){0.f, 0.f, 0.f, 0.f, 0.f, 0.f, 0.f, 0.f};
    v2f ones; ones.x = 1.0f; ones.y = 1.0f;
    for (int k0 = 0; k0 < NUM_CLASSES; k0 += 4) {
        const float2 v = *(const float2*)&myrow[k0 + (hi << 1)];
        v2f a;
        a.x = __expf(v.x - mx);
        a.y = __expf(v.y - mx);
        // D[m][n] += sum_k A[m][k] for all n -> each column of D carries the 16 row sums
        acc = __builtin_amdgcn_wmma_f32_16x16x4_f32(false, a, false, ones,
                                                    (short)0, acc, false, false);
    }

    // ---- Publish row sums: lane 0 holds rows 0..7 (VGPR 0..7), lane 16 rows 8..15 ----
    if (lane == 0) {
        #pragma unroll
        for (int r = 0; r < 8; ++r) rowsum[wave][r] = acc[r];
    } else if (lane == 16) {
        #pragma unroll
        for (int r = 0; r < 8; ++r) rowsum[wave][8 + r] = acc[r];
    }
    __syncthreads();

    // ---- Finalize one row per lane: lse, gather x[target] from resident tile, atomics ----
    if (lane < ROWS_PER_WAVE) {
        const int rLocal = wave * ROWS_PER_WAVE + lane;
        const long long grow = rowBase + rLocal;
        if (grow < n_rows) {
            const float s   = rowsum[wave][lane];
            const float lse = mx + __logf(s);          // lane<16 => lo==lane, mx is this row's max
            const long long t = target[grow];          // reference dtype is int64
            const float xt  = tile[rLocal * ROW_STRIDE + (int)t];
            const float loss = cls_w[(int)t] * (lse - xt);
            atomicAdd(&sums[(int)t], loss);
            atomicAdd(&counts[(int)t], 1.0f);
        }
    }
}

__global__ __launch_bounds__(1024)
void cbce_finalize_kernel(const float* __restrict__ sums,
                          const float* __restrict__ counts,
                          const int* __restrict__ lambda_ptr,
                          float* __restrict__ out, int n_rows)
{
    __shared__ float red[1024];
    const int c = threadIdx.x;
    const float lam = (float)lambda_ptr[0];            // low 32 bits of python int (LE-safe)

    float term = 0.f;
    if (c < NUM_CLASSES) {
        if (lam >= 200.f) {
            term = sums[c];                            // sum(loss)/N branch
        } else {
            const float mean = sums[c] / fmaxf(counts[c], 1.0f);
            const float p = __expf(fminf((mean - 0.5f) / lam, 2.0f));  // BUDGET=0.5, clip=2.0
            term = p * mean;
        }
    }
    red[c] = term;
    __syncthreads();
    for (int s = 512; s > 0; s >>= 1) {
        if (c < s) red[c] += red[c + s];
        __syncthreads();
    }
    if (c == 0) out[0] = red[0] / (float)n_rows;
}

__global__ void zero_ws_kernel(float* __restrict__ ws, int n)
{
    const int i = blockIdx.x * blockDim.x + threadIdx.x;
    if (i < n) ws[i] = 0.f;
}

extern "C" void kernel_launch(void* const* d_in, const int* in_sizes, int n_in,
                              void* d_out, int out_size, void* d_ws, size_t ws_size,
                              hipStream_t stream)
{
    const float*     x      = (const float*)d_in[0];       // [N, 1000] f32
    const long long* target = (const long long*)d_in[1];   // [N] int64
    const float*     cls_w  = (const float*)d_in[2];       // [1000] f32
    const int*       lam    = (const int*)d_in[3];         // scalar
    float* out = (float*)d_out;

    const int n_rows = in_sizes[0] / NUM_CLASSES;

    float* sums   = (float*)d_ws;      // [1024]
    float* counts = sums + 1024;       // [1024]

    zero_ws_kernel<<<2, 1024, 0, stream>>>(sums, 2048);    // ws is poisoned; must re-zero every launch

    const int blocks = (n_rows + ROWS_PER_BLOCK - 1) / ROWS_PER_BLOCK;  // 2048
    cbce_rows_kernel<<<blocks, BLOCK_THREADS, 0, stream>>>(x, target, cls_w, sums, counts, n_rows);

    cbce_finalize_kernel<<<1, 1024, 0, stream>>>(sums, counts, lam, out, n_rows);
}